// VectorQuantizerEMA_2319282339956
// MI455X (gfx1250) — compile-verified
//
#include <hip/hip_runtime.h>
#include <hip/hip_bf16.h>

typedef float v2f __attribute__((ext_vector_type(2)));
typedef float v8f __attribute__((ext_vector_type(8)));

#define B_TOT 65536
#define DIM   256
#define K_NUM 1024
#define DCHUNK 64   // DIM/4
#define NTILE  64   // K_NUM/16

// ---- workspace layout (floats) ----
#define WS_LOSS 0
#define WS_N    1
#define WS_CS   16
#define WS_ESQ  1040
#define WS_DW   2064
#define WS_TOTAL (WS_DW + K_NUM * DIM)

// ---- d_out layout (floats, reference tuple order) ----
#define O_ZQ   ((size_t)0)
#define O_LOSS ((size_t)16777216)
#define O_PERP ((size_t)16777217)
#define O_IDX  ((size_t)16777218)
#define O_EMB  ((size_t)16842754)
#define O_ECS  ((size_t)17104898)
#define O_EMAW ((size_t)17105922)

__global__ __launch_bounds__(256) void vq_zero(float* p, int n) {
    int i = blockIdx.x * 256 + threadIdx.x;
    if (i < n) p[i] = 0.0f;
}

__global__ __launch_bounds__(256) void vq_esq(const float* __restrict__ emb,
                                              float* __restrict__ esq) {
    int k = blockIdx.x * 256 + threadIdx.x;
    if (k < K_NUM) {
        const float* p = emb + (size_t)k * DIM;
        float s = 0.0f;
        for (int d = 0; d < DIM; ++d) s = fmaf(p[d], p[d], s);
        esq[k] = s;
    }
}

// Each wave handles a 16-row M tile; 8 waves/block -> 128 rows/block.
__global__ __launch_bounds__(256) void vq_main(const float* __restrict__ z_e,
                                               const float* __restrict__ emb,
                                               const float* __restrict__ esq,
                                               float* __restrict__ zq_out,
                                               float* __restrict__ idx_out,
                                               float* __restrict__ dw,
                                               float* __restrict__ cs,
                                               float* __restrict__ loss_sum) {
    const int lane = threadIdx.x & 31;
    const int wave = threadIdx.x >> 5;
    const int ln = lane & 15;   // N index within tile / A row
    const int hl = lane >> 4;   // lane half: picks K pair / M half
    const int row_base = blockIdx.x * 128 + wave * 16;

    // Load this wave's entire A tile (16 x 256 f32) into registers.
    // A frag chunk c: v0 = z[m][4c+2h], v1 = z[m][4c+1+2h], m = lane%16, h = lane/16
    v2f a[DCHUNK];
    const float* ap = z_e + (size_t)(row_base + ln) * DIM + 2 * hl;
#pragma unroll
    for (int c = 0; c < DCHUNK; ++c)
        a[c] = *(const v2f*)(ap + 4 * c);

    float best[8];
    int bidx[8];
#pragma unroll
    for (int v = 0; v < 8; ++v) { best[v] = 3.4e38f; bidx[v] = 0x7fffffff; }

    for (int t = 0; t < NTILE; ++t) {
        const float es = esq[t * 16 + ln];
        const float* bp = emb + (size_t)(t * 16 + ln) * DIM + 2 * hl;
        // Two independent accumulator chains so consecutive WMMAs don't
        // serialize on the D->C dependence (issue-rate vs latency).
        v8f acc0 = {0.f, 0.f, 0.f, 0.f, 0.f, 0.f, 0.f, 0.f};
        v8f acc1 = {0.f, 0.f, 0.f, 0.f, 0.f, 0.f, 0.f, 0.f};
#pragma unroll
        for (int c = 0; c < DCHUNK; c += 2) {
            v2f b0 = *(const v2f*)(bp + 4 * c);
            v2f b1 = *(const v2f*)(bp + 4 * (c + 1));
            acc0 = __builtin_amdgcn_wmma_f32_16x16x4_f32(
                false, a[c], false, b0, (short)0, acc0, false, false);
            acc1 = __builtin_amdgcn_wmma_f32_16x16x4_f32(
                false, a[c + 1], false, b1, (short)0, acc1, false, false);
        }
        const int code = t * 16 + ln;
#pragma unroll
        for (int v = 0; v < 8; ++v) {
            float ze = acc0[v] + acc1[v];
            float score = fmaf(ze, -2.0f, es);  // e^2 - 2*z.e
            if (score < best[v]) { best[v] = score; bidx[v] = code; }
        }
    }

    // Reduce over N within each 16-lane half (VGPR v, half h -> row 8h+v).
#pragma unroll
    for (int v = 0; v < 8; ++v) {
#pragma unroll
        for (int off = 1; off < 16; off <<= 1) {
            float ob = __shfl_xor(best[v], off, 32);
            int oi = __shfl_xor(bidx[v], off, 32);
            if (ob < best[v] || (ob == best[v] && oi < bidx[v])) {
                best[v] = ob; bidx[v] = oi;
            }
        }
    }
    if (ln == 0) {
#pragma unroll
        for (int v = 0; v < 8; ++v)
            idx_out[row_base + 8 * hl + v] = (float)bidx[v];
    }

    // Gather z_q rows, loss partial, EMA segment sums.
    float lossAcc = 0.0f;
#pragma unroll
    for (int r = 0; r < 16; ++r) {
        int k = __shfl(bidx[r & 7], (r >> 3) * 16, 32);
        if (lane == 0) unsafeAtomicAdd(cs + k, 1.0f);
        const float* ep = emb + (size_t)k * DIM;
        const float* zp = z_e + (size_t)(row_base + r) * DIM;
        float* op = zq_out + (size_t)(row_base + r) * DIM;
        float* dp = dw + (size_t)k * DIM;
#pragma unroll
        for (int j = 0; j < 8; ++j) {
            int col = lane + 32 * j;
            float e = ep[col];
            float z = zp[col];
            op[col] = e;
            float d = e - z;
            lossAcc = fmaf(d, d, lossAcc);
            unsafeAtomicAdd(dp + col, z);
        }
    }
#pragma unroll
    for (int off = 1; off < 32; off <<= 1) lossAcc += __shfl_xor(lossAcc, off, 32);
    if (lane == 0) unsafeAtomicAdd(loss_sum, lossAcc);
}

__global__ __launch_bounds__(1024) void vq_finalA(const float* __restrict__ cs,
                                                  const float* __restrict__ ema_cs,
                                                  const float* __restrict__ loss_sum,
                                                  float* __restrict__ out_necs,
                                                  float* __restrict__ out_loss,
                                                  float* __restrict__ out_perp,
                                                  float* __restrict__ ws_n) {
    __shared__ float s_n[32], s_e[32];
    int t = threadIdx.x;
    float c = cs[t];
    float necs = 0.99f * ema_cs[t] + 0.01f * c;
    out_necs[t] = necs;
    float apb = c / (float)B_TOT;
    float ent = apb * logf(apb + 1e-10f);
    float n = necs, e = ent;
#pragma unroll
    for (int off = 1; off < 32; off <<= 1) {
        n += __shfl_xor(n, off, 32);
        e += __shfl_xor(e, off, 32);
    }
    int lane = t & 31, w = t >> 5;
    if (lane == 0) { s_n[w] = n; s_e[w] = e; }
    __syncthreads();
    if (w == 0) {
        n = s_n[lane]; e = s_e[lane];
#pragma unroll
        for (int off = 1; off < 32; off <<= 1) {
            n += __shfl_xor(n, off, 32);
            e += __shfl_xor(e, off, 32);
        }
        if (lane == 0) {
            *ws_n = n;
            *out_perp = expf(-e);
            *out_loss = 0.25f * loss_sum[0] / ((float)B_TOT * (float)DIM);
        }
    }
}

__global__ __launch_bounds__(256) void vq_finalB(const float* __restrict__ ema_w,
                                                 const float* __restrict__ dw,
                                                 const float* __restrict__ necs,
                                                 const float* __restrict__ ws_n,
                                                 float* __restrict__ out_emb,
                                                 float* __restrict__ out_ema_w) {
    int k = blockIdx.x;
    int c = threadIdx.x;
    float n = *ws_n;
    float sm = (necs[k] + 1e-5f) / (n + (float)K_NUM * 1e-5f) * n;
    size_t i = (size_t)k * DIM + c;
    float w = 0.99f * ema_w[i] + 0.01f * dw[i];
    out_ema_w[i] = w;
    out_emb[i] = w / sm;
}

extern "C" void kernel_launch(void* const* d_in, const int* in_sizes, int n_in,
                              void* d_out, int out_size, void* d_ws, size_t ws_size,
                              hipStream_t stream) {
    (void)in_sizes; (void)n_in; (void)out_size; (void)ws_size;
    const float* z_e    = (const float*)d_in[0];
    const float* emb    = (const float*)d_in[1];
    const float* ema_cs = (const float*)d_in[2];
    const float* ema_w  = (const float*)d_in[3];
    float* out = (float*)d_out;
    float* ws = (float*)d_ws;

    float* ws_loss = ws + WS_LOSS;
    float* ws_n    = ws + WS_N;
    float* ws_cs   = ws + WS_CS;
    float* ws_esq  = ws + WS_ESQ;
    float* ws_dw   = ws + WS_DW;

    vq_zero<<<(WS_TOTAL + 255) / 256, 256, 0, stream>>>(ws, WS_TOTAL);
    vq_esq<<<(K_NUM + 255) / 256, 256, 0, stream>>>(emb, ws_esq);
    vq_main<<<B_TOT / 128, 256, 0, stream>>>(z_e, emb, ws_esq,
                                             out + O_ZQ, out + O_IDX,
                                             ws_dw, ws_cs, ws_loss);
    vq_finalA<<<1, 1024, 0, stream>>>(ws_cs, ema_cs, ws_loss,
                                      out + O_ECS, out + O_LOSS, out + O_PERP, ws_n);
    vq_finalB<<<K_NUM, 256, 0, stream>>>(ema_w, ws_dw, out + O_ECS, ws_n,
                                         out + O_EMB, out + O_EMAW);
}